// DendriteSingleTimestep_29953101922673
// MI455X (gfx1250) — compile-verified
//
#include <hip/hip_runtime.h>
#include <cstdint>
#include <cstddef>

// ---------------------------------------------------------------------------
// Kobayashi dendrite growth, single explicit timestep.
// phi, tempr : (B, H, W) float32, periodic boundaries.
// out        : phi_new (B*H*W floats) followed by tempr_new (B*H*W floats).
// ---------------------------------------------------------------------------

#define HDIM 2048
#define WDIM 2048

#define TS       32            // output tile edge per workgroup
#define SP       40            // padded LDS row stride (10 float4) for phi/tempr
#define NTHREADS 256           // 8 waves (wave32)
#define QW       (TS + 2)      // prod tile edge (halo 1) = 34

// --- physics constants (folded at compile time) ----------------------------
#define EPSB_C        0.01f
#define KAPPA_C       1.8f
#define DELTA_C       0.02f
#define TEQ_C         1.0f
#define GAMMA_C       10.0f
#define DT_C          1.0e-4f
#define DT_OVER_TAU   0.333333343f          // 1e-4 / 3e-4
#define M_COEF        0.286478907f          // 0.9 / pi
#define INV_DX2       1111.111084f          // 1/(0.03^2)
#define HALF_INV_DX   16.66666603f          // 1/(2*0.03)
#define EAD_COEF      (-0.0012f)            // -EPSILONB*ANISO*DELTA
#define C6T0          0.362357754f          // cos(6*0.2)
#define S6T0          0.932039081f          // sin(6*0.2)

// --- CDNA5 async global->LDS staging (guarded; falls back to plain copy) ---
#if defined(__gfx1250__) && __has_builtin(__builtin_amdgcn_global_load_async_to_lds_b32)
#define USE_ASYNC_LDS 1
#else
#define USE_ASYNC_LDS 0
#endif

#if USE_ASYNC_LDS
typedef int v4i_t __attribute__((vector_size(4 * sizeof(int))));
typedef __attribute__((address_space(1))) int*   gint_ptr;
typedef __attribute__((address_space(3))) int*   lint_ptr;
typedef __attribute__((address_space(1))) v4i_t* gv4i_ptr;
typedef __attribute__((address_space(3))) v4i_t* lv4i_ptr;
#endif

__device__ __forceinline__ void g2l_b32(float* lds_dst, const float* gsrc) {
#if USE_ASYNC_LDS
    __builtin_amdgcn_global_load_async_to_lds_b32(
        (gint_ptr)(uintptr_t)gsrc,
        (lint_ptr)(uint32_t)(uintptr_t)lds_dst,
        0, 0);
#else
    *lds_dst = *gsrc;
#endif
}

__device__ __forceinline__ void g2l_b128(float* lds_dst, const float* gsrc) {
#if USE_ASYNC_LDS && __has_builtin(__builtin_amdgcn_global_load_async_to_lds_b128)
    __builtin_amdgcn_global_load_async_to_lds_b128(
        (gv4i_ptr)(uintptr_t)gsrc,
        (lv4i_ptr)(uint32_t)(uintptr_t)lds_dst,
        0, 0);
#elif USE_ASYNC_LDS
    g2l_b32(lds_dst + 0, gsrc + 0);
    g2l_b32(lds_dst + 1, gsrc + 1);
    g2l_b32(lds_dst + 2, gsrc + 2);
    g2l_b32(lds_dst + 3, gsrc + 3);
#else
    *(float4*)lds_dst = *(const float4*)gsrc;
#endif
}

__device__ __forceinline__ void wait_async_lds() {
#if USE_ASYNC_LDS
#if __has_builtin(__builtin_amdgcn_s_wait_asynccnt)
    __builtin_amdgcn_s_wait_asynccnt(0);
#else
    asm volatile("s_wait_asynccnt 0x0" ::: "memory");
#endif
#endif
}

__device__ __forceinline__ int wrapi(int i, int n) {
    if (i < 0)  i += n;
    if (i >= n) i -= n;
    return i;
}

// LDS layout:
//   s_phi : 36 rows, stride SP. Local (rp, cp), rp,cp in [0,36) ->
//           s_phi[rp*SP + cp + 2]   (LDS col 0 = global col bx*TS - 4)
//   s_tpr : 34 rows, stride SP. Local (rq, cq), rq,cq in [0,34) ->
//           s_tpr[rq*SP + cq + 3]
//   s_p1/s_p2/s_e2 : 34x34, stride QW, index rq*QW + cq.
__global__ __launch_bounds__(NTHREADS)
void dendrite_step_kernel(const float* __restrict__ phi,
                          const float* __restrict__ tempr,
                          float* __restrict__ phi_out,
                          float* __restrict__ tempr_out) {
    __shared__ __attribute__((aligned(16))) float s_phi[36 * SP];  // 1440
    __shared__ __attribute__((aligned(16))) float s_tpr[34 * SP];  // 1360
    __shared__ float s_p1[QW * QW];    // prod_mat1 =  eps*eps' * phidx
    __shared__ float s_p2[QW * QW];    // prod_mat2 = -eps*eps' * phidy
    __shared__ float s_e2[QW * QW];    // eps^2

    const int tid = threadIdx.x;
    const int bx  = blockIdx.x;        // tile col
    const int by  = blockIdx.y;        // tile row
    const int b   = blockIdx.z;        // batch

    const float* __restrict__ phi_b = phi   + (size_t)b * HDIM * WDIM;
    const float* __restrict__ tpr_b = tempr + (size_t)b * HDIM * WDIM;

    const int r0p = by * TS - 2;       // phi tile origin row (halo 2)
    const int c0p = bx * TS - 2;
    const int r0q = by * TS - 1;       // tempr/prod tile origin (halo 1)
    const int c0q = bx * TS - 1;
    const int cal = bx * TS - 4;       // aligned column base of LDS col 0

    if (tid == 0) {
        __builtin_prefetch(phi_b + (size_t)wrapi(r0p, HDIM) * WDIM, 0, 0);
        __builtin_prefetch(tpr_b + (size_t)wrapi(r0q, HDIM) * WDIM, 0, 0);
    }

    const bool fast_cols = (bx > 0) && (bx < (WDIM / TS) - 1);

    if (fast_cols) {
        // ---- vector staging: 10 x b128 per row, columns never wrap ----
        for (int k = tid; k < 36 * 10; k += NTHREADS) {          // phi rows
            int r = k / 10, s = k - r * 10;
            int gr = wrapi(r0p + r, HDIM);
            g2l_b128(&s_phi[r * SP + s * 4],
                     phi_b + (size_t)gr * WDIM + (cal + s * 4));
        }
        for (int k = tid; k < 34 * 10; k += NTHREADS) {          // tempr rows
            int r = k / 10, s = k - r * 10;
            int gr = wrapi(r0q + r, HDIM);
            g2l_b128(&s_tpr[r * SP + s * 4],
                     tpr_b + (size_t)gr * WDIM + (cal + s * 4));
        }
    } else {
        // ---- scalar staging with full periodic wrap (edge columns) ----
        for (int k = tid; k < 36 * 36; k += NTHREADS) {
            int r = k / 36, c = k - r * 36;
            int gr = wrapi(r0p + r, HDIM);
            int gc = wrapi(c0p + c, WDIM);
            g2l_b32(&s_phi[r * SP + c + 2], phi_b + (size_t)gr * WDIM + gc);
        }
        for (int k = tid; k < 34 * 34; k += NTHREADS) {
            int r = k / 34, c = k - r * 34;
            int gr = wrapi(r0q + r, HDIM);
            int gc = wrapi(c0q + c, WDIM);
            g2l_b32(&s_tpr[r * SP + c + 3], tpr_b + (size_t)gr * WDIM + gc);
        }
    }
    wait_async_lds();
    __syncthreads();

    // ---- pass 1: prod_mat1 / prod_mat2 / eps^2 over tile+halo1 (34x34) ----
    // Trig via identities: only one rsqrt per point, rest are FMAs.
    for (int k = tid; k < QW * QW; k += NTHREADS) {
        int rq = k / QW, cq = k - rq * QW;
        int p = (rq + 1) * SP + cq + 3;          // same point in phi tile

        float phidx = (s_phi[p + 1]  - s_phi[p - 1])  * HALF_INV_DX;
        float phidy = (s_phi[p + SP] - s_phi[p - SP]) * HALF_INV_DX;

        float r2 = phidx * phidx + phidy * phidy;
        bool  ok = (r2 > 1e-30f);
        float ri = ok ? rsqrtf(r2) : 0.0f;
        float ct = ok ? phidx * ri : 1.0f;       // cos(theta); theta=0 if grad=0
        float st = phidy * ri;                   // sin(theta) (0 when ri==0)

        // cos/sin of 6*theta by doubling/tripling
        float c2 = ct * ct - st * st;
        float s2 = 2.0f * ct * st;
        float c3 = c2 * ct - s2 * st;
        float s3 = s2 * ct + c2 * st;
        float c6 = c3 * c3 - s3 * s3;
        float s6 = 2.0f * c3 * s3;
        // rotate by -6*theta0
        float cosA = c6 * C6T0 + s6 * S6T0;      // cos(6theta - 6theta0)
        float sinA = s6 * C6T0 - c6 * S6T0;      // sin(6theta - 6theta0)

        float eps  = EPSB_C * (1.0f + DELTA_C * cosA);
        float epsd = EAD_COEF * sinA;            // -EPSILONB*ANISO*DELTA*sinA
        float ee   = eps * epsd;

        s_p1[k] =  ee * phidx;
        s_p2[k] = -ee * phidy;
        s_e2[k] =  eps * eps;
    }
    __syncthreads();

    // ---- pass 2: each thread owns 4 consecutive columns of one row ----
    float* __restrict__ po = phi_out   + (size_t)b * HDIM * WDIM;
    float* __restrict__ to = tempr_out + (size_t)b * HDIM * WDIM;

    const int r  = tid >> 3;           // row 0..31
    const int cb = (tid & 7) << 2;     // column base 0,4,...,28

    float4 pres, tres;
    float* presf = (float*)&pres;
    float* tresf = (float*)&tres;

#pragma unroll
    for (int j = 0; j < 4; ++j) {
        int c = cb + j;
        int p = (r + 2) * SP + c + 4;            // phi center
        int t = (r + 1) * SP + c + 4;            // tempr center
        int q = (r + 1) * QW + (c + 1);          // prod/e2 center

        float phic = s_phi[p];
        float lap_phi = (s_phi[p + 1]  + s_phi[p - 1]  - 2.0f * phic) * INV_DX2
                      + (s_phi[p + SP] + s_phi[p - SP] - 2.0f * phic) * INV_DX2;

        float tc = s_tpr[t];
        float lap_t = (s_tpr[t + 1]  + s_tpr[t - 1]  - 2.0f * tc) * INV_DX2
                    + (s_tpr[t + SP] + s_tpr[t - SP] - 2.0f * tc) * INV_DX2;

        float term1 = (s_p1[q + QW] - s_p1[q - QW]) * HALF_INV_DX;  // d/dy prod1
        float term2 = (s_p2[q + 1]  - s_p2[q - 1])  * HALF_INV_DX;  // d/dx prod2

        float m = M_COEF * atanf(GAMMA_C * (TEQ_C - tc));
        float dphi = DT_OVER_TAU *
            (term1 + term2 + s_e2[q] * lap_phi +
             phic * (1.0f - phic) * (phic - 0.5f + m));

        presf[j] = phic + dphi;
        tresf[j] = tc + DT_C * lap_t + KAPPA_C * dphi;
    }

    size_t g = (size_t)(by * TS + r) * WDIM + (bx * TS + cb);
    *(float4*)(po + g) = pres;
    *(float4*)(to + g) = tres;
}

extern "C" void kernel_launch(void* const* d_in, const int* in_sizes, int n_in,
                              void* d_out, int out_size, void* d_ws, size_t ws_size,
                              hipStream_t stream) {
    (void)n_in; (void)out_size; (void)d_ws; (void)ws_size;
    const float* phi   = (const float*)d_in[0];
    const float* tempr = (const float*)d_in[1];

    const int batch = in_sizes[0] / (HDIM * WDIM);
    float* phi_out   = (float*)d_out;
    float* tempr_out = phi_out + (size_t)batch * HDIM * WDIM;

    dim3 grid(WDIM / TS, HDIM / TS, batch);
    dim3 block(NTHREADS);
    dendrite_step_kernel<<<grid, block, 0, stream>>>(phi, tempr, phi_out, tempr_out);
}